// MultiModalLoss_70540542870061
// MI455X (gfx1250) — compile-verified
//
#include <hip/hip_runtime.h>
#include <hip/hip_bf16.h>
#include <stdint.h>

// Problem constants (from reference)
#define B_  4096
#define M_  6
#define T_  80
#define D_  3
#define ALPHA_ 0.1f
#define LOG2PI_ 1.8378770664093453f

#define NTHREADS 480                 // 15 wave32s; one (m,t) pair per thread
#define SIG_CHUNKS (M_*T_*9*4/16)    // 1080 x 16B
#define MU_CHUNKS  (M_*T_*3*4/16)    // 360 x 16B
#define Y_CHUNKS   (T_*3*4/16)       // 60 x 16B
#define TOTAL_CHUNKS (SIG_CHUNKS + MU_CHUNKS + Y_CHUNKS)  // 1500

typedef __attribute__((ext_vector_type(2))) float v2f;
typedef __attribute__((ext_vector_type(8))) float v8f;

// ---- CDNA5 async global->LDS copy (tracked by ASYNCcnt) -------------------
__device__ __forceinline__ void async_ld_b128(uint32_t lds_byte_off, uint64_t gaddr) {
    asm volatile("global_load_async_to_lds_b128 %0, %1, off"
                 :: "v"(lds_byte_off), "v"(gaddr) : "memory");
}
__device__ __forceinline__ void wait_async0() {
    asm volatile("s_wait_asynccnt 0" ::: "memory");
}
// Generic pointer to LDS byte offset: low 32 bits of a flat pointer into the
// LDS aperture are the wave-relative LDS byte address (ISA 10.2 aperture map).
__device__ __forceinline__ uint32_t lds_off(const void* p) {
    return (uint32_t)(uintptr_t)p;
}

// ---- init: zero the 3 accumulators (l2_sum, ent_sum, gmm_sum) -------------
__global__ void mml_init(float* acc) {
    if (threadIdx.x < 3) acc[threadIdx.x] = 0.0f;
}

// ---- main: one block per batch element b ----------------------------------
__global__ __launch_bounds__(NTHREADS)
void mml_main(const float* __restrict__ mu_g,    // [B,M,T,3]
              const float* __restrict__ conf_g,  // [B,M]
              const float* __restrict__ sig_g,   // [B,M,T,3,3]
              const float* __restrict__ y_g,     // [B,T,3]
              float* __restrict__ acc) {
    __shared__ float s_sig[M_ * T_ * 9];   // 17280 B
    __shared__ float s_mu [M_ * T_ * 3];   //  5760 B
    __shared__ float s_y  [T_ * 3];        //   960 B
    __shared__ float s_red[16 * T_];       // 16 WMMA A-rows x 80 (rows 12..15 zero)
    __shared__ float s_sums[16];

    const int b   = blockIdx.x;
    const int tid = threadIdx.x;

    // ---- Stage this b's slab into LDS via async copies ----
    const uint64_t g_sig = (uint64_t)(sig_g + (size_t)b * (M_*T_*9));
    const uint64_t g_mu  = (uint64_t)(mu_g  + (size_t)b * (M_*T_*3));
    const uint64_t g_y   = (uint64_t)(y_g   + (size_t)b * (T_*3));
    const uint32_t l_sig = lds_off(&s_sig[0]);
    const uint32_t l_mu  = lds_off(&s_mu[0]);
    const uint32_t l_y   = lds_off(&s_y[0]);

    for (int i = tid; i < TOTAL_CHUNKS; i += NTHREADS) {
        uint32_t l; uint64_t g;
        if (i < SIG_CHUNKS) {
            l = l_sig + (uint32_t)i * 16u;
            g = g_sig + (uint64_t)i * 16u;
        } else if (i < SIG_CHUNKS + MU_CHUNKS) {
            uint32_t j = (uint32_t)(i - SIG_CHUNKS);
            l = l_mu + j * 16u;
            g = g_mu + (uint64_t)j * 16u;
        } else {
            uint32_t j = (uint32_t)(i - SIG_CHUNKS - MU_CHUNKS);
            l = l_y + j * 16u;
            g = g_y + (uint64_t)j * 16u;
        }
        async_ld_b128(l, g);
    }

    // Zero the 4 padding rows of the WMMA A staging buffer (branch-free WMMA
    // fetch later; no EXEC toggling inside the WMMA loop).
    if (tid < 4 * T_) s_red[12 * T_ + tid] = 0.0f;

    wait_async0();      // each wave drains its own ASYNCcnt
    __syncthreads();    // then all waves' data is visible in LDS

    // ---- Per-(m,t): 3x3 Cholesky log-prob + squared distance ----
    {
        const int m = tid / T_;
        const int t = tid - m * T_;
        const int e = tid;                       // m*T_ + t
        const float* S = &s_sig[e * 9];
        const float s11 = S[0], s22 = S[4], s33 = S[8];
        const float s21 = S[3], s31 = S[6], s32 = S[7];

        const float d0 = s_y[t*3+0] - s_mu[e*3+0];
        const float d1 = s_y[t*3+1] - s_mu[e*3+1];
        const float d2 = s_y[t*3+2] - s_mu[e*3+2];

        const float l11   = sqrtf(s11);
        const float inv11 = 1.0f / l11;
        const float l21   = s21 * inv11;
        const float l31   = s31 * inv11;
        const float l22sq = s22 - l21 * l21;
        const float l22   = sqrtf(l22sq);
        const float inv22 = 1.0f / l22;
        const float l32   = (s32 - l31 * l21) * inv22;
        const float l33sq = s33 - l31 * l31 - l32 * l32;

        const float z0 = d0 * inv11;
        const float z1 = (d1 - l21 * z0) * inv22;
        const float z2 = (d2 - l31 * z0 - l32 * z1) / sqrtf(l33sq);

        const float quad   = z0*z0 + z1*z1 + z2*z2;
        const float logdet = logf(s11 * l22sq * l33sq);   // = 2*sum(log diag L)
        const float lp     = -0.5f * (3.0f * LOG2PI_ + logdet + quad);
        const float eu     = d0*d0 + d1*d1 + d2*d2;

        s_red[m * T_ + t]        = lp;   // streams 0..5  : log-prob per mode
        s_red[(6 + m) * T_ + t]  = eu;   // streams 6..11 : euclid per mode
    }
    __syncthreads();

    // ---- Wave 0: 12 length-80 reductions via v_wmma_f32_16x16x4_f32 ----
    // A rows = streams (rows 12..15 pre-zeroed), B = ones(4x16):
    // D[r][0] accumulates sum_t stream[r].
    if (tid < 32) {
        const int lane = tid;
        const int row  = lane & 15;
        const int khi  = lane >> 4;              // 0 => K=0,1 ; 1 => K=2,3
        const v2f* arow = (const v2f*)&s_red[row * T_ + khi * 2];  // 8B aligned
        v8f c = {0.f,0.f,0.f,0.f,0.f,0.f,0.f,0.f};
        v2f ones; ones.x = 1.0f; ones.y = 1.0f;  // all-ones B: layout-invariant
        #pragma unroll
        for (int it = 0; it < T_ / 4; ++it) {
            v2f a = arow[it * 2];                // stride 4 floats = 2 v2f
            c = __builtin_amdgcn_wmma_f32_16x16x4_f32(
                    false, a, false, ones, (short)0, c, false, false);
        }
        // D layout: VGPR r, lane 0 -> D[r][0]; lane 16 -> D[8+r][0]
        if (lane == 0) {
            #pragma unroll
            for (int i = 0; i < 8; ++i) s_sums[i] = c[i];       // streams 0..7
        }
        if (lane == 16) {
            #pragma unroll
            for (int i = 0; i < 4; ++i) s_sums[8 + i] = c[i];   // streams 8..11
        }
    }
    __syncthreads();

    // ---- Per-b scalars: mode select, entropy, gmm term; atomic accumulate ----
    if (tid == 0) {
        float lp_sum[M_], eu_sum[M_], cf[M_];
        #pragma unroll
        for (int m = 0; m < M_; ++m) {
            lp_sum[m] = s_sums[m];
            eu_sum[m] = s_sums[6 + m];
            cf[m]     = conf_g[b * M_ + m];
        }
        int best = 0; float bv = eu_sum[0];
        #pragma unroll
        for (int m = 1; m < M_; ++m)
            if (eu_sum[m] < bv) { bv = eu_sum[m]; best = m; }

        // softmax entropy over raw confidences (logits), max-stabilized
        float cmax = cf[0];
        #pragma unroll
        for (int m = 1; m < M_; ++m) cmax = fmaxf(cmax, cf[m]);
        float ex[M_], esum = 0.0f;
        #pragma unroll
        for (int m = 0; m < M_; ++m) { ex[m] = expf(cf[m] - cmax); esum += ex[m]; }
        const float inv = 1.0f / esum;
        float ent = 0.0f;
        #pragma unroll
        for (int m = 0; m < M_; ++m) {
            const float p = ex[m] * inv;
            ent += p * logf(p + 1e-8f);
        }
        const float g = lp_sum[best] + logf(cf[best]);  // gmm_loss = -mean(g)

        atomicAdd(&acc[0], eu_sum[best]);  // l2 numerator (sum over t,d)
        atomicAdd(&acc[1], ent);           // mode_loss = -mean(ent)
        atomicAdd(&acc[2], g);
    }
}

// ---- finalize: 4 scalars ---------------------------------------------------
__global__ void mml_finalize(const float* __restrict__ acc, float* __restrict__ out) {
    const float l2   = acc[0] / (float)((long long)B_ * T_ * D_);
    const float mode = -acc[1] / (float)B_;
    const float gmm  = -acc[2] / (float)B_;
    out[0] = l2 + ALPHA_ * mode;  // total_loss
    out[1] = l2;                  // l2_loss
    out[2] = mode;                // mode_loss
    out[3] = gmm;                 // gmm_loss
}

extern "C" void kernel_launch(void* const* d_in, const int* in_sizes, int n_in,
                              void* d_out, int out_size, void* d_ws, size_t ws_size,
                              hipStream_t stream) {
    const float* mu   = (const float*)d_in[0];  // pred_trajectories [B,M,T,3]
    const float* conf = (const float*)d_in[1];  // pred_confidences  [B,M]
    const float* sig  = (const float*)d_in[2];  // pred_sigmas       [B,M,T,3,3]
    const float* y    = (const float*)d_in[3];  // target_trajectories [B,T,3]
    float* acc = (float*)d_ws;                  // 3 accumulators
    float* out = (float*)d_out;                 // 4 scalars

    mml_init<<<1, 32, 0, stream>>>(acc);
    mml_main<<<B_, NTHREADS, 0, stream>>>(mu, conf, sig, y, acc);
    mml_finalize<<<1, 1, 0, stream>>>(acc, out);
}